// GWOAlignment_56831007261030
// MI455X (gfx1250) — compile-verified
//
#include <hip/hip_runtime.h>
#include <hip/hip_bf16.h>

// ---- problem constants (match reference) ----
#define B_      8
#define N_      16384
#define P_      50
#define SUB_    128
#define NITER_  30
#define ROT_    0.7853981633974483f
#define TRANS_  1.0f

typedef __attribute__((ext_vector_type(2))) float v2f;
typedef __attribute__((ext_vector_type(8))) float v8f;

// ---------------- counter-based RNG (deterministic) ----------------
__device__ __forceinline__ unsigned pcg(unsigned x) {
  x = x * 747796405u + 2891336453u;
  unsigned w = ((x >> ((x >> 28u) + 4u)) ^ x) * 277803737u;
  return (w >> 22u) ^ w;
}
__device__ __forceinline__ float u01(unsigned a, unsigned b, unsigned c) {
  unsigned h = pcg(a + 0x9e3779b9u * pcg(b + 0x85ebca6bu * pcg(c)));
  return (float)(h >> 8) * (1.0f / 16777216.0f);
}

// ---------------- Rodrigues: axis-angle -> 3x3 rotation ----------------
__device__ __forceinline__ void aa_to_R(float ax, float ay, float az, float R[9]) {
  float th  = sqrtf(ax * ax + ay * ay + az * az);
  float inv = 1.0f / fmaxf(th, 1e-8f);
  float x = ax * inv, y = ay * inv, z = az * inv;
  float s = sinf(th), cc = 1.0f - cosf(th);
  R[0] = 1.0f - cc * (y * y + z * z); R[1] = -s * z + cc * x * y; R[2] =  s * y + cc * x * z;
  R[3] =  s * z + cc * x * y;         R[4] = 1.0f - cc * (x * x + z * z); R[5] = -s * x + cc * y * z;
  R[6] = -s * y + cc * x * z;         R[7] =  s * x + cc * y * z; R[8] = 1.0f - cc * (x * x + y * y);
}

// =====================================================================
// Fused GWO kernel: one workgroup per batch, 32 waves, full optimizer
// loop in-kernel. All state in LDS; WMMA f32 16x16x4 chamfer core.
// =====================================================================
__global__ void __launch_bounds__(1024)
k_gwo(const float* __restrict__ src, const float* __restrict__ tgt,
      float* __restrict__ aPosOut) {
  const int b   = blockIdx.x;
  const int tid = threadIdx.x;
  const int ln   = tid & 31;     // lane within wave32
  const int wave = tid >> 5;     // 0..31
  const int half = ln >> 4;      // 0: (x,y)/K=0..1   1: (z,pad)/K=2..3
  const int l15  = ln & 15;

  __shared__ float sS[SUB_ * 3];     // subsampled source (this batch)
  __shared__ float tS[SUB_ * 3];     // subsampled target
  __shared__ float posL[P_ * 6];     // wolf positions
  __shared__ float fitL[P_];         // fitness
  __shared__ float lead[3][6];       // alpha / beta / delta positions
  __shared__ float aFitL;            // alpha fitness

  const float* sg = src + (size_t)b * N_ * 3;
  const float* tg = tgt + (size_t)b * N_ * 3;
  __builtin_prefetch(sg, 0, 0);      // global_prefetch_b8
  __builtin_prefetch(tg, 0, 0);

  // ---- subsample (bijective odd-multiplier index hash) + init positions ----
  if (tid < SUB_) {
    unsigned is = ((unsigned)tid * 2654435761u) & (unsigned)(N_ - 1);
    sS[tid * 3 + 0] = sg[is * 3 + 0];
    sS[tid * 3 + 1] = sg[is * 3 + 1];
    sS[tid * 3 + 2] = sg[is * 3 + 2];
  } else if (tid < 2 * SUB_) {
    int i = tid - SUB_;
    unsigned ix = ((unsigned)i * 40503u + 2257u) & (unsigned)(N_ - 1);
    tS[i * 3 + 0] = tg[ix * 3 + 0];
    tS[i * 3 + 1] = tg[ix * 3 + 1];
    tS[i * 3 + 2] = tg[ix * 3 + 2];
  }
  if (tid < P_ * 6) {
    int p = tid / 6, d = tid % 6;
    float lim = (d < 3) ? ROT_ : TRANS_;
    float u = u01(0x1234567u, (unsigned)(b * P_ + p), (unsigned)d);
    posL[tid] = (p == 0) ? 0.0f : (-lim + 2.0f * lim * u);
  }
  __syncthreads();

  // ---- optimizer loop: it = -1 is the initial eval + leader init ----
  for (int it = -1; it < NITER_; ++it) {
    if (it >= 0) {
      // GWO position update (leader snapshots already in `lead`)
      if (tid < P_ * 6) {
        int p = tid / 6, d = tid % 6;
        float a = 2.0f * (1.0f - (float)it / (float)NITER_);
        float cur = posL[tid];
        unsigned wid = (unsigned)(b * P_ + p);
        float xsum = 0.0f;
#pragma unroll
        for (int L = 0; L < 3; ++L) {
          float l  = lead[L][d];
          float r1 = u01((unsigned)(1000 + it * 6 + L * 2 + 0), wid, (unsigned)d);
          float r2 = u01((unsigned)(1000 + it * 6 + L * 2 + 1), wid, (unsigned)d);
          xsum += l - (2.0f * a * r1 - a) * fabsf(2.0f * r2 * l - cur);
        }
        float x = xsum * (1.0f / 3.0f);
        float lim = (d < 3) ? ROT_ : TRANS_;
        posL[tid] = fminf(fmaxf(x, -lim), lim);
      }
      __syncthreads();
    }

    // ---- chamfer eval: wave-uniform wolf loop (EXEC all-ones for WMMA) ----
    for (int p = wave; p < P_; p += 32) {
      float R[9];
      float aa0 = posL[p * 6 + 0], aa1 = posL[p * 6 + 1], aa2 = posL[p * 6 + 2];
      aa_to_R(aa0, aa1, aa2, R);
      float t0 = posL[p * 6 + 3], t1 = posL[p * 6 + 4], t2 = posL[p * 6 + 5];

      // B tiles (target), |q|^2 folded into K=3 slot
      v2f bt[8];
#pragma unroll
      for (int tj = 0; tj < 8; ++tj) {
        const float* q = &tS[(tj * 16 + l15) * 3];
        float qx = q[0], qy = q[1], qz = q[2];
        float nq = qx * qx + qy * qy + qz * qz;
        bt[tj].x = half ? qz : qx;
        bt[tj].y = half ? nq : qy;
      }

      float colAcc[8];
#pragma unroll
      for (int j = 0; j < 8; ++j) colAcc[j] = 3.4e38f;
      float rowSum = 0.0f;

#pragma unroll 2                       // cap pressure: 32-wave workgroup must be resident
      for (int ti = 0; ti < 8; ++ti) {
        const float* s = &sS[(ti * 16 + l15) * 3];
        float sx = s[0], sy = s[1], sz = s[2];
        float px = R[0] * sx + R[1] * sy + R[2] * sz + t0;
        float py = R[3] * sx + R[4] * sy + R[5] * sz + t1;
        float pz = R[6] * sx + R[7] * sy + R[8] * sz + t2;
        float na = px * px + py * py + pz * pz;
        v2f a;
        a.x = half ? (-2.0f * pz) : (-2.0f * px);
        a.y = half ? 1.0f         : (-2.0f * py);

        float nr[8];
        int base = half ? 8 : 0;
#pragma unroll
        for (int r = 0; r < 8; ++r) nr[r] = __shfl(na, base + r, 32);

        float rowmin[8];
#pragma unroll
        for (int r = 0; r < 8; ++r) rowmin[r] = 3.4e38f;

#pragma unroll
        for (int tj = 0; tj < 8; ++tj) {
          v8f c = {};
          // C = (-2A)·B_aug = -2*dot + |q|^2  =>  d2 = C + |p|^2
          c = __builtin_amdgcn_wmma_f32_16x16x4_f32(false, a, false, bt[tj],
                                                    (short)0, c, false, false);
          float tmin = 3.4e38f;
#pragma unroll
          for (int r = 0; r < 8; ++r) {
            float d2 = c[r] + nr[r];
            rowmin[r] = fminf(rowmin[r], d2);
            tmin = fminf(tmin, d2);
          }
          tmin = fminf(tmin, __shfl_xor(tmin, 16, 32));
          colAcc[tj] = fminf(colAcc[tj], tmin);
        }

#pragma unroll
        for (int r = 0; r < 8; ++r) {
          float v = rowmin[r];
          v = fminf(v, __shfl_xor(v, 1, 32));
          v = fminf(v, __shfl_xor(v, 2, 32));
          v = fminf(v, __shfl_xor(v, 4, 32));
          v = fminf(v, __shfl_xor(v, 8, 32));
          rowSum += v;                 // lanes<16: rows M=r, lanes>=16: rows M=8+r
        }
      }
      rowSum += __shfl_xor(rowSum, 16, 32);

      float colSum = 0.0f;
#pragma unroll
      for (int j = 0; j < 8; ++j) colSum += colAcc[j];
      colSum += __shfl_xor(colSum, 1, 32);
      colSum += __shfl_xor(colSum, 2, 32);
      colSum += __shfl_xor(colSum, 4, 32);
      colSum += __shfl_xor(colSum, 8, 32);

      if (ln == 0) fitL[p] = (rowSum + colSum) * (1.0f / 128.0f);
    }
    __syncthreads();

    // ---- leader selection (top-3 of 50, serial on thread 0) ----
    if (tid == 0) {
      float f0 = 3.4e38f, f1 = 3.4e38f, f2 = 3.4e38f;
      int i0 = 0, i1 = 0, i2 = 0;
      for (int p = 0; p < P_; ++p) {
        float v = fitL[p];
        if (v < f0)      { f2 = f1; i2 = i1; f1 = f0; i1 = i0; f0 = v; i0 = p; }
        else if (v < f1) { f2 = f1; i2 = i1; f1 = v;  i1 = p; }
        else if (v < f2) { f2 = v;  i2 = p; }
      }
      if (it < 0 || f0 < aFitL) {
        aFitL = f0;
        for (int d = 0; d < 6; ++d) lead[0][d] = posL[i0 * 6 + d];
      }
      for (int d = 0; d < 6; ++d) {
        lead[1][d] = posL[i1 * 6 + d];
        lead[2][d] = posL[i2 * 6 + d];
      }
    }
    __syncthreads();
  }

  if (tid < 6) aPosOut[b * 6 + tid] = lead[0][tid];
}

// ---------------- final: R_best, t_best, aligned cloud ----------------
__global__ void k_final(const float* __restrict__ source, const float* __restrict__ aPos,
                        float* __restrict__ out) {
  int g = blockIdx.x * blockDim.x + threadIdx.x;
  if (g >= B_ * N_) return;
  int b = g / N_, n = g % N_;
  const float* ap = aPos + b * 6;
  float R[9];
  aa_to_R(ap[0], ap[1], ap[2], R);
  float t0 = ap[3], t1 = ap[4], t2 = ap[5];
  const float* s = source + (size_t)g * 3;
  float sx = s[0], sy = s[1], sz = s[2];
  float* al = out + 96;                       // after R(72) + t(24)
  al[(size_t)g * 3 + 0] = R[0] * sx + R[1] * sy + R[2] * sz + t0;
  al[(size_t)g * 3 + 1] = R[3] * sx + R[4] * sy + R[5] * sz + t1;
  al[(size_t)g * 3 + 2] = R[6] * sx + R[7] * sy + R[8] * sz + t2;
  if (n == 0) {
    for (int k = 0; k < 9; ++k) out[b * 9 + k] = R[k];
    out[72 + b * 3 + 0] = t0;
    out[72 + b * 3 + 1] = t1;
    out[72 + b * 3 + 2] = t2;
  }
}

extern "C" void kernel_launch(void* const* d_in, const int* in_sizes, int n_in,
                              void* d_out, int out_size, void* d_ws, size_t ws_size,
                              hipStream_t stream) {
  const float* source = (const float*)d_in[0];
  const float* target = (const float*)d_in[1];
  float* aPosW = (float*)d_ws;                // B*6 = 48 floats
  float* out   = (float*)d_out;

  // Entire optimizer in one launch: 8 workgroups (one WGP each), 32 waves,
  // all state LDS-resident, barriers between phases.
  k_gwo<<<dim3(B_), dim3(1024), 0, stream>>>(source, target, aPosW);

  // Bandwidth-trivial final transform (1.5 MB in / 1.5 MB out).
  k_final<<<dim3((B_ * N_ + 255) / 256), dim3(256), 0, stream>>>(source, aPosW, out);
}